// ScaleDotProductAttention_88132728914540
// MI455X (gfx1250) — compile-verified
//
#include <hip/hip_runtime.h>

#define BH   128
#define SEQ  512
#define HD   64
#define NEGV (-1e12f)
#define SCL  (0.125f)      // 1/sqrt(64)

typedef float v2f __attribute__((ext_vector_type(2)));
typedef float v8f __attribute__((ext_vector_type(8)));

#define QSTR 68            // padded LDS row strides (floats) -> conflict-free
#define KSTR 68
#define PSTR 516

__global__ __launch_bounds__(128)
void sdpa_wmma_f32_kernel(const float* __restrict__ Q, const float* __restrict__ K,
                          const float* __restrict__ V, const float* __restrict__ SEL,
                          const int* __restrict__ MASK,
                          float* __restrict__ CTX, float* __restrict__ ATT)
{
    __shared__ float sQ[16 * QSTR];          //  4.3 KB
    __shared__ float sK[4][16 * KSTR];       // 17.4 KB (per-wave K tiles)
    __shared__ float sP[16 * PSTR];          // 33.0 KB (scores -> probs)

    const int qb    = blockIdx.x;            // 0..31  (q block of 16 rows)
    const int bh    = blockIdx.y;            // 0..127 (batch*head)
    const int tid   = threadIdx.x;           // 0..127
    const int wave  = tid >> 5;
    const int lane  = tid & 31;
    const int hlf   = lane >> 4;             // 0 = lanes 0-15, 1 = lanes 16-31
    const int ln16  = lane & 15;
    const int qbase = qb * 16;
    const int koff  = hlf * 2;               // WMMA f32 A/B half-lane K offset

    const float* Qp = Q   + ((size_t)bh * SEQ + qbase) * HD;
    const float* Kp = K   + (size_t)bh * SEQ * HD;
    const float* Vp = V   + (size_t)bh * SEQ * HD;
    const float* Sp = SEL + ((size_t)bh * SEQ + qbase) * SEQ;
    const int*   Mp = MASK + ((size_t)bh * SEQ + qbase) * SEQ;

    // ---- stage Q tile (16 x 64) into padded LDS ----
    #pragma unroll
    for (int i = 0; i < 2; ++i) {
        int off = (tid + i * 128) * 4;       // dword offset inside contiguous tile
        int row = off >> 6, col = off & 63;
        const float4 d4 = *(const float4*)(Qp + off);
        float* dst = &sQ[row * QSTR + col];
        dst[0] = d4.x; dst[1] = d4.y; dst[2] = d4.z; dst[3] = d4.w;
    }
    __syncthreads();

    // ---- S = scale * Q K^T : each wave computes 8 of 32 16x16 tiles ----
    for (int t = 0; t < 8; ++t) {
        const int kbase = (wave * 8 + t) * 16;

        // stage this wave's K tile (16x64, contiguous) with coalesced float4
        #pragma unroll
        for (int i = 0; i < 8; ++i) {
            int off = (lane + i * 32) * 4;
            int row = off >> 6, col = off & 63;
            const float4 d4 = *(const float4*)(Kp + (size_t)kbase * HD + off);
            float* dst = &sK[wave][row * KSTR + col];
            dst[0] = d4.x; dst[1] = d4.y; dst[2] = d4.z; dst[3] = d4.w;
        }
        // producer and consumer are the same wave: LDS-count wait, no barrier
        asm volatile("s_wait_dscnt 0" ::: "memory");

        v8f acc = {0.f, 0.f, 0.f, 0.f, 0.f, 0.f, 0.f, 0.f};
        #pragma unroll
        for (int c = 0; c < 16; ++c) {       // K = 64 in chunks of 4
            const int cc = c * 4 + koff;
            v2f a, b;
            a[0] = sQ[ln16 * QSTR + cc];
            a[1] = sQ[ln16 * QSTR + cc + 1];
            b[0] = sK[wave][ln16 * KSTR + cc];      // B[d][n] = K[kbase+n][d]
            b[1] = sK[wave][ln16 * KSTR + cc + 1];
            acc = __builtin_amdgcn_wmma_f32_16x16x4_f32(false, a, false, b,
                                                        (short)0, acc, false, false);
        }
        // scatter scaled raw scores: VGPR j -> row j (+8 for upper half-lanes)
        #pragma unroll
        for (int j = 0; j < 8; ++j) {
            int r = j + hlf * 8;
            sP[r * PSTR + kbase + ln16] = acc[j] * SCL;
        }
    }
    __syncthreads();

    // ---- softmax rows (selector + mask fused here, coalesced streams) ----
    for (int rr = 0; rr < 4; ++rr) {
        const int r = wave * 4 + rr;
        const float* selRow = Sp + (size_t)r * SEQ;
        const int*   mRow   = Mp + (size_t)r * SEQ;
        float vals[16];
        float m = NEGV;
        #pragma unroll
        for (int i = 0; i < 16; ++i) {
            int col = lane + i * 32;
            float s = sP[r * PSTR + col] + selRow[col];
            s = (mRow[col] != 0) ? s : NEGV;
            vals[i] = s;
            m = fmaxf(m, s);
        }
        #pragma unroll
        for (int o = 16; o > 0; o >>= 1)
            m = fmaxf(m, __shfl_xor(m, o, 32));
        float sum = 0.f;
        #pragma unroll
        for (int i = 0; i < 16; ++i) {
            float e = __expf(vals[i] - m);
            vals[i] = e;
            sum += e;
        }
        #pragma unroll
        for (int o = 16; o > 0; o >>= 1)
            sum += __shfl_xor(sum, o, 32);
        const float inv = 1.0f / sum;
        float* aRow = ATT + ((size_t)bh * SEQ + qbase + r) * SEQ;
        #pragma unroll
        for (int i = 0; i < 16; ++i) {
            int col = lane + i * 32;
            float p = vals[i] * inv;
            sP[r * PSTR + col] = p;          // keep probs for P @ V
            aRow[col] = p;                   // stream attn out, coalesced
        }
    }
    __syncthreads();

    // ---- context = P(16x512) @ V(512x64): wave owns one 16-wide D slice ----
    const int dbase = wave * 16;
    v8f acc = {0.f, 0.f, 0.f, 0.f, 0.f, 0.f, 0.f, 0.f};
    #pragma unroll 4
    for (int c = 0; c < 128; ++c) {          // K = 512 in chunks of 4
        const int cc = c * 4 + koff;
        v2f a, b;
        a[0] = sP[ln16 * PSTR + cc];
        a[1] = sP[ln16 * PSTR + cc + 1];
        b[0] = Vp[(size_t)cc * HD + dbase + ln16];        // B[k][n] = V[k][dbase+n]
        b[1] = Vp[(size_t)(cc + 1) * HD + dbase + ln16];
        acc = __builtin_amdgcn_wmma_f32_16x16x4_f32(false, a, false, b,
                                                    (short)0, acc, false, false);
    }
    #pragma unroll
    for (int j = 0; j < 8; ++j) {
        int r = j + hlf * 8;
        CTX[((size_t)bh * SEQ + qbase + r) * HD + dbase + ln16] = acc[j];
    }
}

extern "C" void kernel_launch(void* const* d_in, const int* in_sizes, int n_in,
                              void* d_out, int out_size, void* d_ws, size_t ws_size,
                              hipStream_t stream) {
    (void)in_sizes; (void)n_in; (void)out_size; (void)d_ws; (void)ws_size;
    const float* q    = (const float*)d_in[0];
    const float* k    = (const float*)d_in[1];
    const float* v    = (const float*)d_in[2];
    const float* sel  = (const float*)d_in[3];
    const int*   mask = (const int*)d_in[4];

    float* ctx = (float*)d_out;                                   // [BH, SEQ, HD]
    float* att = (float*)d_out + (size_t)BH * SEQ * HD;           // [BH, SEQ, SEQ]

    dim3 grid(SEQ / 16, BH);
    dim3 block(128);
    sdpa_wmma_f32_kernel<<<grid, block, 0, stream>>>(q, k, v, sel, mask, ctx, att);
}